// Gaussian_32890859553362
// MI455X (gfx1250) — compile-verified
//
#include <hip/hip_runtime.h>
#include <math.h>

// Problem constants (match reference)
#define BATCH 4
#define FMAP  64
#define NPTS  4096
#define NELEM (BATCH * FMAP * NPTS)   // 1,048,576

typedef __attribute__((ext_vector_type(2))) float v2f;
typedef __attribute__((ext_vector_type(8))) float v8f;

// ---------------------------------------------------------------------------
// Stage 1: per-block partial sums (sum, sum of squares) in double.
// 1024 blocks x 256 threads x 4 elements = 1,048,576 elements exactly.
// ---------------------------------------------------------------------------
__global__ __launch_bounds__(256)
void gauss_reduce_partial(const float* __restrict__ emb,
                          double* __restrict__ partials) {
  __shared__ double s_sum[256];
  __shared__ double s_sq[256];
  const int t = threadIdx.x;
  const size_t base = (size_t)blockIdx.x * 1024;
  double s = 0.0, q = 0.0;
#pragma unroll
  for (int k = 0; k < 4; ++k) {
    const float v = emb[base + (size_t)k * 256 + t];
    s += (double)v;
    q += (double)v * (double)v;
  }
  s_sum[t] = s;
  s_sq[t]  = q;
  __syncthreads();
  for (int off = 128; off > 0; off >>= 1) {
    if (t < off) {
      s_sum[t] += s_sum[t + off];
      s_sq[t]  += s_sq[t + off];
    }
    __syncthreads();
  }
  if (t == 0) {
    partials[blockIdx.x]        = s_sum[0];
    partials[1024 + blockIdx.x] = s_sq[0];
  }
}

// ---------------------------------------------------------------------------
// Stage 2: finalize std (Bessel ddof=1) -> stats[0] = 1/std^2
// Single block, deterministic tree reduction of 1024 partials.
// ---------------------------------------------------------------------------
__global__ __launch_bounds__(256)
void gauss_finalize(const double* __restrict__ partials,
                    float* __restrict__ stats) {
  __shared__ double s_sum[256];
  __shared__ double s_sq[256];
  const int t = threadIdx.x;
  double s = 0.0, q = 0.0;
#pragma unroll
  for (int k = 0; k < 4; ++k) {
    s += partials[k * 256 + t];
    q += partials[1024 + k * 256 + t];
  }
  s_sum[t] = s;
  s_sq[t]  = q;
  __syncthreads();
  for (int off = 128; off > 0; off >>= 1) {
    if (t < off) {
      s_sum[t] += s_sum[t + off];
      s_sq[t]  += s_sq[t + off];
    }
    __syncthreads();
  }
  if (t == 0) {
    const double n    = (double)NELEM;
    const double sum  = s_sum[0];
    const double ssq  = s_sq[0];
    const double var  = (ssq - sum * sum / n) / (n - 1.0);  // ddof=1
    const double istd = 1.0 / sqrt(var);
    stats[0] = (float)(istd * istd);  // 1/std^2
    stats[1] = (float)istd;           // 1/std
  }
}

// ---------------------------------------------------------------------------
// Stage 3: sq[b,n] = invstd^2 * sum_f emb[b,f,n]^2   (B*N = 16384 values)
// ---------------------------------------------------------------------------
__global__ __launch_bounds__(256)
void gauss_sq(const float* __restrict__ emb,
              const float* __restrict__ stats,
              float* __restrict__ sqout) {
  const int idx = blockIdx.x * 256 + threadIdx.x;   // 0 .. B*N-1
  const int b = idx >> 12;                          // / NPTS
  const int n = idx & (NPTS - 1);
  const float* p = emb + (size_t)b * FMAP * NPTS + n;
  float acc = 0.0f;
#pragma unroll
  for (int f = 0; f < FMAP; ++f) {
    const float v = p[(size_t)f * NPTS];
    acc = fmaf(v, v, acc);
  }
  sqout[idx] = acc * stats[0];
}

// ---------------------------------------------------------------------------
// Stage 4: Gram matrix via V_WMMA_F32_16X16X4_F32, register-blocked 2x4
// tiles per wave (32n x 64m), + Gaussian epilogue.
//
// f32 16x4 A-fragment layout (ISA 7.12.2): lanes 0-15 hold M=lane for K=0(v0),
// K=1(v1); lanes 16-31 hold M=lane-16 for K=2(v0), K=3(v1). B (4x16) mirrored
// over N. C/D: VGPR r, lane -> (M = r + 8*(lane>>4), N = lane&15).
//
// Register blocking: per k-step, 4 A loads + 8 B loads feed 8 WMMAs
// (vs 4 loads per WMMA unblocked) -> VMEM issue rate drops below the
// 256MB NT-store HBM floor (~11us @ 23.3 TB/s).
//
// We accumulate raw emb.emb (invstd^2 folded into the epilogue), then
//   d2 = max(0, (sq_n + sq_m - 2*invstd2*acc) / F),  out = exp(-d2/2)
// streamed with non-temporal stores (write-once, > 192MB L2).
// ---------------------------------------------------------------------------
__global__ __launch_bounds__(256)
void gauss_wmma(const float* __restrict__ emb,
                const float* __restrict__ sqin,
                const float* __restrict__ stats,
                float* __restrict__ out) {
  const int lane = threadIdx.x & 31;
  const int wid  = threadIdx.x >> 5;
  const int half = lane >> 4;    // 0: lanes 0-15, 1: lanes 16-31
  const int row  = lane & 15;

  const int b  = blockIdx.z;
  // 8 waves/block = 4 n-groups x 2 m-groups; block tile = 128n x 128m
  const int n0 = (blockIdx.y << 7) + ((wid >> 1) << 5);   // wave: 32 rows (2 tiles)
  const int m0 = (blockIdx.x << 7) + ((wid & 1) << 6);    // wave: 64 cols (4 tiles)

  const float* bb = emb + (size_t)b * (size_t)FMAP * NPTS;
  const int cA = n0 + row;   // A columns: cA, cA+16
  const int cB = m0 + row;   // B columns: cB, cB+16, cB+32, cB+48

  v8f acc[2][4] = {};
#pragma unroll
  for (int kk = 0; kk < 16; ++kk) {
    const int f0 = 4 * kk + 2 * half;
    const float* p0 = bb + (size_t)f0 * NPTS;        // K row f0
    const float* p1 = p0 + NPTS;                     // K row f0+1
    v2f a[2], bv[4];
#pragma unroll
    for (int i = 0; i < 2; ++i) {
      a[i].x = p0[cA + 16 * i];
      a[i].y = p1[cA + 16 * i];
    }
#pragma unroll
    for (int j = 0; j < 4; ++j) {
      bv[j].x = p0[cB + 16 * j];
      bv[j].y = p1[cB + 16 * j];
    }
#pragma unroll
    for (int i = 0; i < 2; ++i)
#pragma unroll
      for (int j = 0; j < 4; ++j)
        acc[i][j] = __builtin_amdgcn_wmma_f32_16x16x4_f32(
            false, a[i], false, bv[j], (short)0, acc[i][j], false, false);
  }

  const float invstd2 = stats[0];
  const float inv_f   = 1.0f / (float)FMAP;

  float sqm[4];
#pragma unroll
  for (int j = 0; j < 4; ++j) sqm[j] = sqin[(b << 12) + cB + 16 * j];

#pragma unroll
  for (int i = 0; i < 2; ++i) {
    const int nbase = n0 + 16 * i + half * 8;
#pragma unroll
    for (int r = 0; r < 8; ++r) {
      const float sqn = sqin[(b << 12) + nbase + r];
      float* orow = out + ((size_t)b * NPTS + (size_t)(nbase + r)) * NPTS
                        + cB;
#pragma unroll
      for (int j = 0; j < 4; ++j) {
        float d2 = (sqn + sqm[j] - 2.0f * invstd2 * acc[i][j][r]) * inv_f;
        d2 = fmaxf(d2, 0.0f);
        const float val = __expf(-0.5f * d2);   // v_exp_f32
        __builtin_nontemporal_store(val, orow + 16 * j);
      }
    }
  }
}

// ---------------------------------------------------------------------------
// Host-side launcher
// ---------------------------------------------------------------------------
extern "C" void kernel_launch(void* const* d_in, const int* in_sizes, int n_in,
                              void* d_out, int out_size, void* d_ws, size_t ws_size,
                              hipStream_t stream) {
  (void)in_sizes; (void)n_in; (void)out_size; (void)ws_size;

  // Inputs in setup_inputs() order: [0]=adj_in (unused), [1]=emb_in
  const float* emb = (const float*)d_in[1];
  float* out = (float*)d_out;

  // Workspace layout (~82 KB):
  //   [0      .. 16384)  double partials[2048]  (sum[1024], sumsq[1024])
  //   [16384  .. 16448)  float  stats[16]       (invstd2, invstd, ...)
  //   [16448  .. 81984)  float  sq[B*N]
  char* ws = (char*)d_ws;
  double* partials = (double*)ws;
  float*  stats    = (float*)(ws + 16384);
  float*  sqbuf    = (float*)(ws + 16448);

  gauss_reduce_partial<<<1024, 256, 0, stream>>>(emb, partials);
  gauss_finalize<<<1, 256, 0, stream>>>(partials, stats);
  gauss_sq<<<(BATCH * NPTS) / 256, 256, 0, stream>>>(emb, stats, sqbuf);

  // Each wave: 2x4 tiles of 16x16 -> block (8 waves) covers 128x128.
  dim3 grid(NPTS / 128, NPTS / 128, BATCH);  // (32, 32, 4)
  gauss_wmma<<<grid, 256, 0, stream>>>(emb, sqbuf, stats, out);
}